// Whiten_22892175688444
// MI455X (gfx1250) — compile-verified
//
#include <hip/hip_runtime.h>

typedef float v2f __attribute__((ext_vector_type(2)));
typedef float v4f __attribute__((ext_vector_type(4)));
typedef float v8f __attribute__((ext_vector_type(8)));

#define NFEAT 64
#define DDIM  9216
#define CHUNK 64
#define LSTR  68   // LDS row stride (floats): 4*r mod 64 -> 16 distinct banks over 16 rows

__global__ __launch_bounds__(256)
void whiten_kernel(const float* __restrict__ X, float* __restrict__ Out) {
    __shared__ float sX[NFEAT][LSTR];   // streamed data chunk
    __shared__ float sG[NFEAT][LSTR];   // Gram -> cov -> L
    __shared__ float sW[NFEAT][LSTR];   // inv(L)
    __shared__ float sSum[NFEAT][4];
    __shared__ float sMean[NFEAT];

    const int b    = blockIdx.x;
    const int t    = threadIdx.x;
    const int wave = t >> 5;
    const int lane = t & 31;

    const float* Xb = X   + (size_t)b * NFEAT * DDIM;
    float*       Ob = Out + (size_t)b * NFEAT * DDIM;

    // tile assignment: wave w -> row block ti = w>>1, col blocks tj0, tj0+1
    const int ti    = wave >> 1;
    const int tj0   = (wave & 1) * 2;
    const int l15   = lane & 15;
    const int khalf = (lane >> 4) * 2;       // lanes 0-15 -> K,K+1 ; lanes 16-31 -> K+2,K+3
    const int mrow  = ti * 16 + l15;         // A-fragment row (M = lane)
    const int crow0 = tj0 * 16 + l15;        // B row for Gram tile 0 (N = lane)
    const int crow1 = crow0 + 16;            // B row for Gram tile 1

    v8f acc0 = {}; v8f acc1 = {};
    float rsum = 0.f;
    const int srow = t >> 2;
    const int scol = (t & 3) * 16;

    // ---------------- Pass 1: row sums + Gram = X * X^T via WMMA ----------------
    for (int d0 = 0; d0 < DDIM; d0 += CHUNK) {
        #pragma unroll
        for (int i = 0; i < 4; ++i) {
            int idx = t + i * 256;           // float4 index 0..1023
            int row = idx >> 4;
            int c4  = (idx & 15) << 2;
            v4f v = *(const v4f*)(Xb + (size_t)row * DDIM + d0 + c4);
            *(v4f*)&sX[row][c4] = v;
        }
        if (d0 + CHUNK < DDIM)
            __builtin_prefetch(Xb + (size_t)srow * DDIM + d0 + CHUNK + scol, 0, 0);
        __syncthreads();

        // per-row partial sums (4 threads per row)
        float s = 0.f;
        #pragma unroll
        for (int c = 0; c < 16; ++c) s += sX[srow][scol + c];
        rsum += s;

        // Gram tiles: D[M][N] += sum_K A[M][K]*B[K][N], A,B both rows of X
        #pragma unroll
        for (int kk = 0; kk < CHUNK; kk += 4) {
            int kb = kk + khalf;
            v2f a, b0, b1;
            a.x  = sX[mrow][kb];  a.y  = sX[mrow][kb + 1];
            b0.x = sX[crow0][kb]; b0.y = sX[crow0][kb + 1];
            b1.x = sX[crow1][kb]; b1.y = sX[crow1][kb + 1];
            acc0 = __builtin_amdgcn_wmma_f32_16x16x4_f32(false, a, false, b0,
                                                         (short)0, acc0, false, false);
            acc1 = __builtin_amdgcn_wmma_f32_16x16x4_f32(false, a, false, b1,
                                                         (short)0, acc1, false, false);
        }
        __syncthreads();
    }

    // spill Gram tiles + row sums to LDS
    sSum[srow][t & 3] = rsum;
    #pragma unroll
    for (int v = 0; v < 8; ++v) {
        int row = ti * 16 + v + ((lane >> 4) << 3);   // C layout: VGPR v, M = v (+8 for hi lanes)
        int col = tj0 * 16 + l15;
        sG[row][col]      = acc0[v];
        sG[row][col + 16] = acc1[v];
    }
    __syncthreads();

    if (t < NFEAT) {
        float S = sSum[t][0] + sSum[t][1] + sSum[t][2] + sSum[t][3];
        sSum[t][0] = S;
        sMean[t]   = S * (1.0f / DDIM);
    }
    __syncthreads();

    // cov[r][c] = (G[r][c] - S_r*S_c/D) / (D-1)
    {
        const float invD   = 1.0f / DDIM;
        const float invDm1 = 1.0f / (DDIM - 1);
        #pragma unroll
        for (int i = 0; i < 16; ++i) {
            int idx = t + i * 256;
            int r = idx >> 6, c = idx & 63;
            sG[r][c] = (sG[r][c] - sSum[r][0] * sSum[c][0] * invD) * invDm1;
        }
    }
    __syncthreads();

    // ---------------- Cholesky (right-looking, in-place in sG) ----------------
    for (int j = 0; j < NFEAT; ++j) {
        if (t == 0) sG[j][j] = sqrtf(sG[j][j]);
        __syncthreads();
        float Ljj = sG[j][j];
        if (t > j && t < NFEAT) sG[t][j] /= Ljj;
        __syncthreads();
        if (t > j && t < NFEAT) {
            float lij = sG[t][j];
            for (int k = j + 1; k <= t; ++k) sG[t][k] -= lij * sG[k][j];
        }
        __syncthreads();
    }

    // ---------------- W = inv(L): one thread per column, forward substitution ----
    #pragma unroll
    for (int i = 0; i < 16; ++i) { int idx = t + i * 256; sW[idx >> 6][idx & 63] = 0.f; }
    __syncthreads();
    if (t < NFEAT) {
        const int c = t;
        sW[c][c] = 1.0f / sG[c][c];
        for (int i = c + 1; i < NFEAT; ++i) {
            float s = 0.f;
            for (int k = c; k < i; ++k) s += sG[i][k] * sW[k][c];
            sW[i][c] = -s / sG[i][i];
        }
    }
    __syncthreads();

    // hoist A-fragments of W (loop-invariant over D)
    v2f aW[16];
    #pragma unroll
    for (int kk = 0; kk < CHUNK; kk += 4) {
        int kb = kk + khalf;
        aW[kk >> 2].x = sW[mrow][kb];
        aW[kk >> 2].y = sW[mrow][kb + 1];
    }

    // ---------------- Pass 2: out = W * (X - mean) via WMMA ----------------
    const int nc0 = tj0 * 16 + l15;
    for (int d0 = 0; d0 < DDIM; d0 += CHUNK) {
        #pragma unroll
        for (int i = 0; i < 4; ++i) {
            int idx = t + i * 256;
            int row = idx >> 4;
            int c4  = (idx & 15) << 2;
            v4f v = *(const v4f*)(Xb + (size_t)row * DDIM + d0 + c4);
            float m = sMean[row];
            v.x -= m; v.y -= m; v.z -= m; v.w -= m;
            *(v4f*)&sX[row][c4] = v;
        }
        __syncthreads();

        v8f o0 = {}, o1 = {};
        #pragma unroll
        for (int kk = 0; kk < CHUNK; kk += 4) {
            int kb = kk + khalf;
            v2f b0, b1;
            b0.x = sX[kb][nc0];      b0.y = sX[kb + 1][nc0];
            b1.x = sX[kb][nc0 + 16]; b1.y = sX[kb + 1][nc0 + 16];
            o0 = __builtin_amdgcn_wmma_f32_16x16x4_f32(false, aW[kk >> 2], false, b0,
                                                       (short)0, o0, false, false);
            o1 = __builtin_amdgcn_wmma_f32_16x16x4_f32(false, aW[kk >> 2], false, b1,
                                                       (short)0, o1, false, false);
        }

        #pragma unroll
        for (int v = 0; v < 8; ++v) {
            int row = ti * 16 + v + ((lane >> 4) << 3);
            size_t base = (size_t)row * DDIM + d0 + nc0;
            Ob[base]      = o0[v];
            Ob[base + 16] = o1[v];
        }
        __syncthreads();
    }
}

extern "C" void kernel_launch(void* const* d_in, const int* in_sizes, int n_in,
                              void* d_out, int out_size, void* d_ws, size_t ws_size,
                              hipStream_t stream) {
    (void)in_sizes; (void)n_in; (void)d_ws; (void)ws_size; (void)out_size;
    const float* X = (const float*)d_in[0];
    float* O = (float*)d_out;
    whiten_kernel<<<64, 256, 0, stream>>>(X, O);
}